// CenterLoss_19481971655234
// MI455X (gfx1250) — compile-verified
//
#include <hip/hip_runtime.h>

typedef __attribute__((ext_vector_type(2))) float v2f;
typedef __attribute__((ext_vector_type(8))) float v8f;

// One wave32 handles 16 samples: D[16][16] = X_tile(16xK) * C_gathered(Kx16),
// accumulated K=4 at a time with V_WMMA_F32_16X16X4_F32; only diag(D) is used.
__global__ __launch_bounds__(256) void center_loss_wmma_kernel(
    const float* __restrict__ x,
    const int*   __restrict__ labels,
    const float* __restrict__ centers,
    float*       __restrict__ wave_sums)
{
    constexpr int D = 512;
    const int lane = threadIdx.x & 31;
    const int wave = (int)((blockIdx.x * blockDim.x + threadIdx.x) >> 5);
    const int s0   = wave * 16;          // first sample of this wave's tile
    const int r    = lane & 15;          // row/col 0..15 within tile
    const int half = lane >> 4;          // 0: K%4 in {0,1}, 1: K%4 in {2,3}

    const int lab = labels[s0 + r];      // gathered center row for column r

    // A-matrix 16x4 f32 layout: lanes 0-15 hold K0(v0),K1(v1); lanes 16-31 K2,K3.
    // B-matrix 4x16 mirrors with N = lane&15. Both reduce to base + k + 2*half.
    const float* __restrict__ xrow = x       + (long)(s0 + r) * D + 2 * half;
    const float* __restrict__ crow = centers + (long)lab      * D + 2 * half;

    v8f acc0 = {}, acc1 = {}, acc2 = {}, acc3 = {};
    float xn = 0.0f, cn = 0.0f;

#pragma unroll 4
    for (int k = 0; k < D; k += 16) {
        v2f a0, a1, a2, a3, b0, b1, b2, b3;
        a0.x = xrow[k + 0];  a0.y = xrow[k + 1];
        a1.x = xrow[k + 4];  a1.y = xrow[k + 5];
        a2.x = xrow[k + 8];  a2.y = xrow[k + 9];
        a3.x = xrow[k + 12]; a3.y = xrow[k + 13];
        b0.x = crow[k + 0];  b0.y = crow[k + 1];
        b1.x = crow[k + 4];  b1.y = crow[k + 5];
        b2.x = crow[k + 8];  b2.y = crow[k + 9];
        b3.x = crow[k + 12]; b3.y = crow[k + 13];

        // squared-norm partials ride along on VALU (co-executes with WMMA)
        xn += a0.x*a0.x + a0.y*a0.y + a1.x*a1.x + a1.y*a1.y
            + a2.x*a2.x + a2.y*a2.y + a3.x*a3.x + a3.y*a3.y;
        cn += b0.x*b0.x + b0.y*b0.y + b1.x*b1.x + b1.y*b1.y
            + b2.x*b2.x + b2.y*b2.y + b3.x*b3.x + b3.y*b3.y;

        // 8 args: (neg_a, A, neg_b, B, c_mod, C, reuse_a, reuse_b)
        acc0 = __builtin_amdgcn_wmma_f32_16x16x4_f32(false, a0, false, b0, (short)0, acc0, false, false);
        acc1 = __builtin_amdgcn_wmma_f32_16x16x4_f32(false, a1, false, b1, (short)0, acc1, false, false);
        acc2 = __builtin_amdgcn_wmma_f32_16x16x4_f32(false, a2, false, b2, (short)0, acc2, false, false);
        acc3 = __builtin_amdgcn_wmma_f32_16x16x4_f32(false, a3, false, b3, (short)0, acc3, false, false);
    }

    v8f acc = acc0 + acc1 + acc2 + acc3;

    // combine K-half norm partials: row r's full norm = lane r + lane r+16
    xn += __shfl_xor(xn, 16, 32);
    cn += __shfl_xor(cn, 16, 32);

    // Diagonal of 16x16 f32 C/D layout:
    //   (M=r,N=r): r<8 -> lane r, elem r ; r>=8 -> lane r+16, elem r-8
    const int dv  = (lane < 16) ? lane : (lane - 24);
    const int sel = dv & 7;
    float e0 = acc[0], e1 = acc[1], e2 = acc[2], e3 = acc[3];
    float e4 = acc[4], e5 = acc[5], e6 = acc[6], e7 = acc[7];
    float m0 = (sel & 1) ? e1 : e0;
    float m1 = (sel & 1) ? e3 : e2;
    float m2 = (sel & 1) ? e5 : e4;
    float m3 = (sel & 1) ? e7 : e6;
    float n0 = (sel & 2) ? m1 : m0;
    float n1 = (sel & 2) ? m3 : m2;
    float dot = (sel & 4) ? n1 : n0;

    // rows 8..15 live on lanes 24..31; pull them to lanes 8..15
    float dot_hi = __shfl_xor(dot, 16, 32);
    if (half == 0 && r >= 8) dot = dot_hi;

    float val = 0.0f;
    if (half == 0) {   // lanes 0..15 now each own one sample's pieces
        float dist = xn + cn - 2.0f * dot;
        dist = fminf(fmaxf(dist, 1e-12f), 1e12f);
        val = dist;
    }

    // wave reduction (lanes >=16 contribute 0)
    val += __shfl_xor(val, 16, 32);
    val += __shfl_xor(val,  8, 32);
    val += __shfl_xor(val,  4, 32);
    val += __shfl_xor(val,  2, 32);
    val += __shfl_xor(val,  1, 32);

    if (lane == 0) wave_sums[wave] = val;
}

// Deterministic final reduction (no float atomics) + masked-clamp bias term.
__global__ __launch_bounds__(256) void center_loss_reduce_kernel(
    const float* __restrict__ wave_sums, int nwaves, float bias,
    float* __restrict__ out)
{
    __shared__ float sm[256];
    const int t = threadIdx.x;
    float s = 0.0f;
    for (int i = t; i < nwaves; i += 256) s += wave_sums[i];
    sm[t] = s;
    __syncthreads();
    for (int stride = 128; stride > 0; stride >>= 1) {
        if (t < stride) sm[t] += sm[t + stride];
        __syncthreads();
    }
    if (t == 0) out[0] = sm[0] + bias;
}

extern "C" void kernel_launch(void* const* d_in, const int* in_sizes, int n_in,
                              void* d_out, int out_size, void* d_ws, size_t ws_size,
                              hipStream_t stream) {
    const float* x       = (const float*)d_in[0];   // (16,256,512) f32
    const int*   labels  = (const int*)  d_in[1];   // (16,256) int
    const float* centers = (const float*)d_in[2];   // (10000,512) f32
    float* out       = (float*)d_out;
    float* wave_sums = (float*)d_ws;

    const int nsamp      = in_sizes[1];         // 4096
    const int numClasses = in_sizes[2] / 512;   // 10000
    const int nwaves     = nsamp / 16;          // 256 (16 samples per wave)
    const int blocks     = nwaves / 8;          // 8 waves (256 threads) per block

    // C-1 masked entries per sample each clamp up to 1e-12
    const float bias = (float)((double)nsamp * (double)(numClasses - 1) * 1e-12);

    center_loss_wmma_kernel<<<blocks, 256, 0, stream>>>(x, labels, centers, wave_sums);
    center_loss_reduce_kernel<<<1, 256, 0, stream>>>(wave_sums, nwaves, bias, out);
}